// D2RLCritic_64304250356442
// MI455X (gfx1250) — compile-verified
//
#include <hip/hip_runtime.h>
#include <hip/hip_bf16.h>

typedef __attribute__((ext_vector_type(2))) float v2f;
typedef __attribute__((ext_vector_type(8))) float v8f;

#define NN 100000
#define NE 1600000
#define NG 256
#define FIN 64
#define TILES 6250   // NN / 16

// ---------------------------------------------------------------------------
// Layer-1 node GEMMs: xl = x @ w1l, xr = x @ w1r   (100k x 64 @ 64 x 16)
// One wave32 per 16-row tile, V_WMMA_F32_16X16X4_F32, K stepped by 4.
// ---------------------------------------------------------------------------
__global__ void __launch_bounds__(256) linear1_wmma(
    const float* __restrict__ x, const float* __restrict__ w1l,
    const float* __restrict__ w1r, float* __restrict__ xl,
    float* __restrict__ xr) {
  __shared__ float sWl[FIN * 16];
  __shared__ float sWr[FIN * 16];
  int t = threadIdx.x;
  for (int i = t; i < FIN * 16; i += 256) { sWl[i] = w1l[i]; sWr[i] = w1r[i]; }
  __syncthreads();

  int wave = t >> 5, lane = t & 31;
  int tile = blockIdx.x * 8 + wave;
  if (tile >= TILES) return;                 // wave-uniform: EXEC stays all-1s
  int m = lane & 15, half = lane >> 4;       // A-row == B/C-col for this lane
  const float* arow = x + (size_t)(tile * 16 + m) * FIN;

  v8f cl = {}; v8f cr = {};
  #pragma unroll
  for (int kk = 0; kk < FIN / 4; ++kk) {
    int k0 = kk * 4 + half * 2;
    v2f a;  a.x  = arow[k0];            a.y  = arow[k0 + 1];
    v2f bl; bl.x = sWl[k0 * 16 + m];    bl.y = sWl[(k0 + 1) * 16 + m];
    v2f br; br.x = sWr[k0 * 16 + m];    br.y = sWr[(k0 + 1) * 16 + m];
    cl = __builtin_amdgcn_wmma_f32_16x16x4_f32(false, a, false, bl, (short)0, cl, false, false);
    cr = __builtin_amdgcn_wmma_f32_16x16x4_f32(false, a, false, br, (short)0, cr, false, false);
  }
  #pragma unroll
  for (int r = 0; r < 8; ++r) {
    size_t row = (size_t)tile * 16 + r + half * 8;   // C layout: vgpr r -> M=r / M=r+8
    xl[row * 16 + m] = cl[r];
    xr[row * 16 + m] = cr[r];
  }
}

// ---------------------------------------------------------------------------
// Layer-2 node GEMMs with BN folded in: a = h1*s + t ; yl = a@w2l, yr = a@w2r
// ---------------------------------------------------------------------------
__global__ void __launch_bounds__(256) linear2_wmma(
    const float* __restrict__ h1, const float* __restrict__ sv,
    const float* __restrict__ tv, const float* __restrict__ w2l,
    const float* __restrict__ w2r, float* __restrict__ yl,
    float* __restrict__ yr) {
  __shared__ float sWl[16 * 16];
  __shared__ float sWr[16 * 16];
  __shared__ float sS[16], sT[16];
  int t = threadIdx.x;
  if (t < 256) { sWl[t] = w2l[t]; sWr[t] = w2r[t]; }
  if (t < 16)  { sS[t] = sv[t]; sT[t] = tv[t]; }
  __syncthreads();

  int wave = t >> 5, lane = t & 31;
  int tile = blockIdx.x * 8 + wave;
  if (tile >= TILES) return;
  int m = lane & 15, half = lane >> 4;
  const float* arow = h1 + (size_t)(tile * 16 + m) * 16;

  v8f cl = {}; v8f cr = {};
  #pragma unroll
  for (int kk = 0; kk < 4; ++kk) {
    int k0 = kk * 4 + half * 2;
    v2f a;  a.x  = arow[k0] * sS[k0] + sT[k0];
            a.y  = arow[k0 + 1] * sS[k0 + 1] + sT[k0 + 1];
    v2f bl; bl.x = sWl[k0 * 16 + m];  bl.y = sWl[(k0 + 1) * 16 + m];
    v2f br; br.x = sWr[k0 * 16 + m];  br.y = sWr[(k0 + 1) * 16 + m];
    cl = __builtin_amdgcn_wmma_f32_16x16x4_f32(false, a, false, bl, (short)0, cl, false, false);
    cr = __builtin_amdgcn_wmma_f32_16x16x4_f32(false, a, false, br, (short)0, cr, false, false);
  }
  #pragma unroll
  for (int r = 0; r < 8; ++r) {
    size_t row = (size_t)tile * 16 + r + half * 8;
    yl[row * 16 + m] = cl[r];
    yr[row * 16 + m] = cr[r];
  }
}

// ---------------------------------------------------------------------------
// Edge scatter-add of 16-dim features (+ optional degree count)
// ---------------------------------------------------------------------------
__global__ void __launch_bounds__(256) edge_agg(
    const int* __restrict__ src, const int* __restrict__ dst,
    const float* __restrict__ feat, float* __restrict__ agg,
    float* __restrict__ cnt) {
  int e = blockIdx.x * 256 + threadIdx.x;
  if (e >= NE) return;
  int s = src[e], d = dst[e];
  const float4* v = (const float4*)(feat + (size_t)s * 16);
  float* o = agg + (size_t)d * 16;
  #pragma unroll
  for (int q = 0; q < 4; ++q) {
    float4 f = v[q];
    atomicAdd(o + q * 4 + 0, f.x);
    atomicAdd(o + q * 4 + 1, f.y);
    atomicAdd(o + q * 4 + 2, f.z);
    atomicAdd(o + q * 4 + 3, f.w);
  }
  if (cnt) atomicAdd(cnt + d, 1.0f);
}

// ---------------------------------------------------------------------------
// h1 = relu(agg/max(cnt,1) + b1l + xr); accumulate per-channel sum / sumsq
// ---------------------------------------------------------------------------
__global__ void __launch_bounds__(256) h1_stats(
    const float* __restrict__ agg, const float* __restrict__ cnt,
    const float* __restrict__ xr, const float* __restrict__ b1l,
    float* __restrict__ h1, float* __restrict__ colsum,
    float* __restrict__ colsq) {
  __shared__ float ssum[16], ssq[16];
  int t = threadIdx.x;
  if (t < 16) { ssum[t] = 0.0f; ssq[t] = 0.0f; }
  __syncthreads();
  int n = blockIdx.x * 256 + t;
  if (n < NN) {
    float ic = 1.0f / fmaxf(cnt[n], 1.0f);
    #pragma unroll
    for (int c = 0; c < 16; ++c) {
      float h = fmaxf(agg[(size_t)n * 16 + c] * ic + b1l[c] + xr[(size_t)n * 16 + c], 0.0f);
      h1[(size_t)n * 16 + c] = h;
      atomicAdd(&ssum[c], h);
      atomicAdd(&ssq[c], h * h);
    }
  }
  __syncthreads();
  if (t < 16) { atomicAdd(&colsum[t], ssum[t]); atomicAdd(&colsq[t], ssq[t]); }
}

__global__ void finalize_stats(const float* __restrict__ colsum,
                               const float* __restrict__ colsq,
                               const float* __restrict__ g1,
                               const float* __restrict__ be1,
                               float* __restrict__ sv, float* __restrict__ tv) {
  int c = threadIdx.x;
  if (c < 16) {
    float mu  = colsum[c] * (1.0f / (float)NN);
    float var = colsq[c] * (1.0f / (float)NN) - mu * mu;
    float sc  = g1[c] * rsqrtf(var + 1e-5f);
    sv[c] = sc;
    tv[c] = be1[c] - mu * sc;
  }
}

// ---------------------------------------------------------------------------
// h2 = relu(agg/max(cnt,1) + b2l + yr); pool per-graph via atomics
// ---------------------------------------------------------------------------
__global__ void __launch_bounds__(256) h2_enc(
    const float* __restrict__ agg, const float* __restrict__ cnt,
    const float* __restrict__ yr, const float* __restrict__ b2l,
    const int* __restrict__ batch, float* __restrict__ enc_sum,
    float* __restrict__ enc_cnt) {
  int n = blockIdx.x * 256 + threadIdx.x;
  if (n >= NN) return;
  float ic = 1.0f / fmaxf(cnt[n], 1.0f);
  int g = batch[n];
  #pragma unroll
  for (int c = 0; c < 16; ++c) {
    float h = fmaxf(agg[(size_t)n * 16 + c] * ic + b2l[c] + yr[(size_t)n * 16 + c], 0.0f);
    atomicAdd(&enc_sum[g * 16 + c], h);
  }
  atomicAdd(&enc_cnt[g], 1.0f);
}

// ---------------------------------------------------------------------------
// Head: one 256-thread block, one graph per thread, BN via LDS reductions.
// ---------------------------------------------------------------------------
__device__ __forceinline__ void bn_fold(const float* vals, int nch,
                                        const float* gamma, const float* beta,
                                        float* red, float* sS, float* sT) {
  int t = threadIdx.x;
  if (t < 64) red[t] = 0.0f;
  __syncthreads();
  for (int c = 0; c < nch; ++c) {
    atomicAdd(&red[c], vals[c]);
    atomicAdd(&red[32 + c], vals[c] * vals[c]);
  }
  __syncthreads();
  if (t < nch) {
    float mu  = red[t] * (1.0f / (float)NG);
    float var = red[32 + t] * (1.0f / (float)NG) - mu * mu;
    float sc  = gamma[t] * rsqrtf(var + 1e-5f);
    sS[t] = sc;
    sT[t] = beta[t] - mu * sc;
  }
  __syncthreads();
}

__global__ void __launch_bounds__(256) head_kernel(
    const float* __restrict__ enc_sum, const float* __restrict__ enc_cnt,
    const float* __restrict__ gl1, const float* __restrict__ bl1,
    const float* __restrict__ W1, const float* __restrict__ bW1,
    const float* __restrict__ gl2, const float* __restrict__ bl2,
    const float* __restrict__ W2, const float* __restrict__ bW2,
    const float* __restrict__ gl3, const float* __restrict__ bl3,
    const float* __restrict__ W3, const float* __restrict__ bW3,
    const float* __restrict__ Wf, const float* __restrict__ bWf,
    float* __restrict__ out) {
  __shared__ float red[64];
  __shared__ float sS[32], sT[32];
  int g = threadIdx.x;

  float xe[16];
  float ic = 1.0f / fmaxf(enc_cnt[g], 1.0f);
  #pragma unroll
  for (int c = 0; c < 16; ++c) xe[c] = enc_sum[g * 16 + c] * ic;

  // z1 = relu(bn(xe) @ W1 + bW1)
  bn_fold(xe, 16, gl1, bl1, red, sS, sT);
  float z1[16];
  #pragma unroll
  for (int j = 0; j < 16; ++j) {
    float acc = bW1[j];
    for (int c = 0; c < 16; ++c) acc += (xe[c] * sS[c] + sT[c]) * W1[c * 16 + j];
    z1[j] = fmaxf(acc, 0.0f);
  }

  // z2 = relu(bn([z1, xe]) @ W2 + bW2)
  float c2[32];
  #pragma unroll
  for (int c = 0; c < 16; ++c) { c2[c] = z1[c]; c2[16 + c] = xe[c]; }
  __syncthreads();
  bn_fold(c2, 32, gl2, bl2, red, sS, sT);
  float z2[16];
  #pragma unroll
  for (int j = 0; j < 16; ++j) {
    float acc = bW2[j];
    for (int c = 0; c < 32; ++c) acc += (c2[c] * sS[c] + sT[c]) * W2[c * 16 + j];
    z2[j] = fmaxf(acc, 0.0f);
  }

  // z3 = relu(bn([z2, xe]) @ W3 + bW3)
  float c3[32];
  #pragma unroll
  for (int c = 0; c < 16; ++c) { c3[c] = z2[c]; c3[16 + c] = xe[c]; }
  __syncthreads();
  bn_fold(c3, 32, gl3, bl3, red, sS, sT);
  float z3[16];
  #pragma unroll
  for (int j = 0; j < 16; ++j) {
    float acc = bW3[j];
    for (int c = 0; c < 32; ++c) acc += (c3[c] * sS[c] + sT[c]) * W3[c * 16 + j];
    z3[j] = fmaxf(acc, 0.0f);
  }

  float o = bWf[0];
  #pragma unroll
  for (int j = 0; j < 16; ++j) o += z3[j] * Wf[j];
  out[g] = o;
}

// ---------------------------------------------------------------------------
extern "C" void kernel_launch(void* const* d_in, const int* in_sizes, int n_in,
                              void* d_out, int out_size, void* d_ws, size_t ws_size,
                              hipStream_t stream) {
  (void)in_sizes; (void)n_in; (void)out_size; (void)ws_size;
  const float* x     = (const float*)d_in[0];
  const int*   ei    = (const int*)d_in[1];
  const int*   src   = ei;
  const int*   dst   = ei + NE;
  const int*   batch = (const int*)d_in[2];
  const float* w1l = (const float*)d_in[3];
  const float* b1l = (const float*)d_in[4];
  const float* w1r = (const float*)d_in[5];
  const float* g1  = (const float*)d_in[6];
  const float* be1 = (const float*)d_in[7];
  const float* w2l = (const float*)d_in[8];
  const float* b2l = (const float*)d_in[9];
  const float* w2r = (const float*)d_in[10];
  const float* gl1 = (const float*)d_in[11];
  const float* bl1 = (const float*)d_in[12];
  const float* W1  = (const float*)d_in[13];
  const float* bW1 = (const float*)d_in[14];
  const float* gl2 = (const float*)d_in[15];
  const float* bl2 = (const float*)d_in[16];
  const float* W2  = (const float*)d_in[17];
  const float* bW2 = (const float*)d_in[18];
  const float* gl3 = (const float*)d_in[19];
  const float* bl3 = (const float*)d_in[20];
  const float* W3  = (const float*)d_in[21];
  const float* bW3 = (const float*)d_in[22];
  const float* Wf  = (const float*)d_in[23];
  const float* bWf = (const float*)d_in[24];

  float* ws      = (float*)d_ws;
  float* xl      = ws;                       // also yl after layer 1 retires
  float* xr      = ws + (size_t)NN * 16;     // also yr
  float* agg     = ws + (size_t)2 * NN * 16; // reused for both layers
  float* h1      = ws + (size_t)3 * NN * 16;
  float* cnt     = ws + (size_t)4 * NN * 16;
  float* colsum  = cnt + NN;
  float* colsq   = colsum + 16;
  float* sv      = colsq + 16;
  float* tv      = sv + 16;
  float* enc_sum = tv + 16;
  float* enc_cnt = enc_sum + NG * 16;

  const int EB = (NE + 255) / 256;   // 6250
  const int NB = (NN + 255) / 256;   // 391
  const int GB = (TILES + 7) / 8;    // 782 blocks x 8 waves

  // zero all atomic accumulators every call (graph-replay safe)
  hipMemsetAsync(agg, 0, (size_t)NN * 16 * sizeof(float), stream);
  hipMemsetAsync(cnt, 0, (size_t)(NN + 64 + NG * 16 + NG) * sizeof(float), stream);

  linear1_wmma<<<GB, 256, 0, stream>>>(x, w1l, w1r, xl, xr);
  edge_agg<<<EB, 256, 0, stream>>>(src, dst, xl, agg, cnt);
  h1_stats<<<NB, 256, 0, stream>>>(agg, cnt, xr, b1l, h1, colsum, colsq);
  finalize_stats<<<1, 32, 0, stream>>>(colsum, colsq, g1, be1, sv, tv);
  linear2_wmma<<<GB, 256, 0, stream>>>(h1, sv, tv, w2l, w2r, xl, xr);
  hipMemsetAsync(agg, 0, (size_t)NN * 16 * sizeof(float), stream);
  edge_agg<<<EB, 256, 0, stream>>>(src, dst, xl, agg, nullptr);
  h2_enc<<<NB, 256, 0, stream>>>(agg, cnt, xr, b2l, batch, enc_sum, enc_cnt);
  head_kernel<<<1, 256, 0, stream>>>(enc_sum, enc_cnt, gl1, bl1, W1, bW1,
                                     gl2, bl2, W2, bW2, gl3, bl3, W3, bW3,
                                     Wf, bWf, (float*)d_out);
}